// BracketNet_62758062129263
// MI455X (gfx1250) — compile-verified
//
#include <hip/hip_runtime.h>

typedef __attribute__((ext_vector_type(2))) float v2f;
typedef __attribute__((ext_vector_type(8))) float v8f;

#define S_LEN 2048
#define B_SZ  64
#define D_SZ  512
#define H_N   8
#define DIM   64
#define K2    128
#define CSTR  68   // LDS row stride (floats) for ctx tiles, padded vs 64 banks

// ---------------------------------------------------------------------------
// Phase 1 (fully parallel): Z[t,b,h,n] = bias[h,n] + sum_k W[h,n,64+k]*x[t,b,h,k]
// Result staged into d_out (same shape as final output); phase 2 overwrites it.
// ---------------------------------------------------------------------------
__global__ __launch_bounds__(256) void bracket_phase1(
    const float* __restrict__ src, const float* __restrict__ W,
    const float* __restrict__ bias, float* __restrict__ out)
{
  const int lane = threadIdx.x & 31;
  const int wv   = threadIdx.x >> 5;        // 8 waves per block
  const int h    = blockIdx.y;
  const int row0 = (blockIdx.x * 8 + wv) * 16;   // row index into [S*B)
  const int lh   = lane >> 4;               // half-wave select (k offset)
  const int ml   = lane & 15;               // A: M row / B: N col

  // A-fragments: x rows row0..row0+15, k = 0..63 (held in VGPRs, reused 4x)
  v2f a[16];
  const float* xbase = src + (size_t)(row0 + ml) * D_SZ + h * DIM;
#pragma unroll
  for (int kk = 0; kk < 16; ++kk) {
    const int kb = 4 * kk + 2 * lh;
    a[kk] = *(const v2f*)(xbase + kb);
  }

  const float* wb = W + (size_t)h * DIM * K2;     // [64 n][128 k]
#pragma unroll
  for (int nt = 0; nt < 4; ++nt) {
    const int n = nt * 16 + ml;
    v8f c = {};
#pragma unroll
    for (int kk = 0; kk < 16; ++kk) {
      const int kb = 64 + 4 * kk + 2 * lh;        // x-half of W
      const v2f bf = *(const v2f*)(wb + (size_t)n * K2 + kb);
      c = __builtin_amdgcn_wmma_f32_16x16x4_f32(false, a[kk], false, bf,
                                                (short)0, c, false, false);
    }
    const float bv = bias[h * DIM + n];
#pragma unroll
    for (int r = 0; r < 8; ++r) {
      const int m = r + 8 * lh;
      out[(size_t)(row0 + m) * D_SZ + h * DIM + n] = c[r] + bv;
    }
  }
}

// ---------------------------------------------------------------------------
// Phase 2 (serial scan): ctx = GELU(W_a @ ctx + Z[t]);  out[t] = x[t] + ctx
// One block per (head, 16-batch group); 4 waves = 4 N-tiles of 16.
// W_a fragments pinned in VGPRs for the whole scan; ctx double-buffered in LDS.
// K=64 accumulation split into TWO independent WMMA chains to halve the
// dependent-latency depth on the serial critical path.
// ---------------------------------------------------------------------------
__global__ __launch_bounds__(128) void bracket_phase2(
    const float* __restrict__ src, const float* __restrict__ W,
    float* __restrict__ out)
{
  __shared__ float ctxs[2][16 * CSTR];

  const int lane = threadIdx.x & 31;
  const int wv   = threadIdx.x >> 5;        // 4 waves
  const int h    = blockIdx.y;
  const int b0   = blockIdx.x * 16;
  const int lh   = lane >> 4;
  const int ml   = lane & 15;
  const int n    = wv * 16 + ml;

  // Preload W_a (k = 0..63) B-fragments into registers: 16 x v2f
  v2f wa[16];
  const float* wrow = W + (size_t)(h * DIM + n) * K2;
#pragma unroll
  for (int kk = 0; kk < 16; ++kk)
    wa[kk] = *(const v2f*)(wrow + 4 * kk + 2 * lh);

  // ctx buffer 0 := 0 (initialize_context)
  for (int i = threadIdx.x; i < 16 * CSTR; i += 128) ctxs[0][i] = 0.0f;
  __syncthreads();

#pragma unroll 1
  for (int t = 0; t < S_LEN; ++t) {
    const int    p     = t & 1;
    const size_t rbase = ((size_t)t * B_SZ + b0) * D_SZ + h * DIM;

    // Chain 0 accumulator initialized from Z[t] (phase-1 result in d_out);
    // chain 1 starts at zero so its WMMAs can issue before the Z loads land.
    v8f c0;
#pragma unroll
    for (int r = 0; r < 8; ++r) {
      const int m = r + 8 * lh;
      c0[r] = out[rbase + (size_t)m * D_SZ + n];
    }
    v8f c1 = {};

    // Hint-prefetch next step's x / Z rows while the WMMA chains run
    if (t + 1 < S_LEN) {
      __builtin_prefetch(src + rbase + (size_t)B_SZ * D_SZ + (size_t)ml * D_SZ, 0, 0);
      __builtin_prefetch(out + rbase + (size_t)B_SZ * D_SZ + (size_t)ml * D_SZ, 0, 0);
    }

    // c0 += W_a[:, 0:32] @ ctx[0:32] ;  c1 += W_a[:, 32:64] @ ctx[32:64]
    // Two independent 8-deep accumulate chains instead of one 16-deep chain.
#pragma unroll
    for (int kk = 0; kk < 8; ++kk) {
      const int kb0 = 4 * kk + 2 * lh;            // k in [0, 32)
      const int kb1 = 32 + 4 * kk + 2 * lh;       // k in [32, 64)
      const v2f af0 = *(const v2f*)&ctxs[p][ml * CSTR + kb0];
      const v2f af1 = *(const v2f*)&ctxs[p][ml * CSTR + kb1];
      c1 = __builtin_amdgcn_wmma_f32_16x16x4_f32(false, af1, false, wa[kk + 8],
                                                 (short)0, c1, false, false);
      c0 = __builtin_amdgcn_wmma_f32_16x16x4_f32(false, af0, false, wa[kk],
                                                 (short)0, c0, false, false);
    }

    // Exact-erf GELU, residual add, store; stage new ctx into the other buffer
#pragma unroll
    for (int r = 0; r < 8; ++r) {
      const int   m  = r + 8 * lh;
      const float y  = c0[r] + c1[r];
      const float g  = 0.5f * y * (1.0f + erff(y * 0.70710678118654752f));
      const float xv = src[rbase + (size_t)m * D_SZ + n];
      out[rbase + (size_t)m * D_SZ + n] = xv + g;
      ctxs[p ^ 1][m * CSTR + n] = g;
    }
    __syncthreads();   // new ctx visible to all waves before next step
  }
}

// ---------------------------------------------------------------------------
extern "C" void kernel_launch(void* const* d_in, const int* in_sizes, int n_in,
                              void* d_out, int out_size, void* d_ws, size_t ws_size,
                              hipStream_t stream) {
  (void)in_sizes; (void)n_in; (void)out_size; (void)d_ws; (void)ws_size;
  const float* src  = (const float*)d_in[0];   // [S, B, D] f32
  const float* W    = (const float*)d_in[1];   // [H, DIM, 2*DIM] f32
  const float* bias = (const float*)d_in[2];   // [H, DIM] f32
  float* out = (float*)d_out;                  // [S, B, D] f32

  dim3 g1(S_LEN * B_SZ / (16 * 8), H_N);       // 1024 x 8 blocks, 8 waves each
  bracket_phase1<<<g1, 256, 0, stream>>>(src, W, bias, out);

  dim3 g2(B_SZ / 16, H_N);                     // 4 x 8 blocks, 4 waves each
  bracket_phase2<<<g2, 128, 0, stream>>>(src, W, out);
}